// ArcticMoE_26130581029431
// MI455X (gfx1250) — compile-verified
//
#include <hip/hip_runtime.h>
#include <math.h>

// Problem dims (fixed by reference)
#define T_TOK 2048
#define HID   2048
#define INTER 1024
#define NE    8

// Tiling
#define BM 128      // token rows per block
#define BK 32       // K-slice per LDS stage
#define SX 36       // LDS row stride in floats (16B-aligned rows, conflict-free frags)
#define TILE_F (BM * SX)                  // floats per tile buffer
#define SMEM_BYTES (4 * TILE_F * 4)       // xs0, bs0, xs1, bs1

typedef __attribute__((ext_vector_type(2))) float v2f;
typedef __attribute__((ext_vector_type(8))) float v8f;

static __device__ __forceinline__ v8f zero8() {
    v8f z = {0.f, 0.f, 0.f, 0.f, 0.f, 0.f, 0.f, 0.f};
    return z;
}

// CDNA5 async memory->LDS copy, tracked by ASYNCcnt (no VGPR round trip).
#define ASYNC_B128(ldsa, ga)                                                   \
    asm volatile("global_load_async_to_lds_b128 %0, %1, off"                   \
                 :: "v"(ldsa), "v"(ga) : "memory")
#define WAIT_ASYNC_8()  asm volatile("s_wait_asynccnt 0x8" ::: "memory")
#define WAIT_ASYNC_0()  asm volatile("s_wait_asynccnt 0x0" ::: "memory")

static __device__ __forceinline__ unsigned lds_addr(const void* p) {
    // Generic (flat) LDS pointer: low 32 bits are the wave-relative LDS offset.
    return (unsigned)(uintptr_t)p;
}

// ---------------------------------------------------------------------------
// Router: logits = x @ gate_w.T ; softmax ; top-2 ; renorm ; dense cw[T, E]
// ---------------------------------------------------------------------------
__global__ __launch_bounds__(256) void moe_router_kernel(
    const float* __restrict__ x, const float* __restrict__ gw,
    float* __restrict__ cw)
{
    const int t    = blockIdx.x * 8 + (threadIdx.x >> 5);
    const int lane = threadIdx.x & 31;

    float acc[NE];
#pragma unroll
    for (int e = 0; e < NE; ++e) acc[e] = 0.f;

    const float* xrow = x + (size_t)t * HID;
    for (int j = lane; j < HID; j += 32) {
        const float xv = xrow[j];
#pragma unroll
        for (int e = 0; e < NE; ++e) acc[e] += xv * gw[e * HID + j];
    }
#pragma unroll
    for (int e = 0; e < NE; ++e) {
        float v = acc[e];
#pragma unroll
        for (int m = 16; m >= 1; m >>= 1) v += __shfl_xor(v, m, 32);
        acc[e] = v;
    }

    if (lane == 0) {
        float mx = acc[0];
#pragma unroll
        for (int e = 1; e < NE; ++e) mx = fmaxf(mx, acc[e]);
        float p[NE], sum = 0.f;
#pragma unroll
        for (int e = 0; e < NE; ++e) { p[e] = __expf(acc[e] - mx); sum += p[e]; }
        const float inv = 1.f / sum;
#pragma unroll
        for (int e = 0; e < NE; ++e) p[e] *= inv;

        int i1 = 0;
#pragma unroll
        for (int e = 1; e < NE; ++e) if (p[e] > p[i1]) i1 = e;
        int i2 = (i1 == 0) ? 1 : 0;
#pragma unroll
        for (int e = 0; e < NE; ++e) if (e != i1 && p[e] > p[i2]) i2 = e;

        const float s = p[i1] + p[i2];
        float w[NE];
#pragma unroll
        for (int e = 0; e < NE; ++e) w[e] = 0.f;
        w[i1] = p[i1] / s;
        w[i2] = p[i2] / s;
#pragma unroll
        for (int e = 0; e < NE; ++e) cw[t * NE + e] = w[e];
    }
}

// ---------------------------------------------------------------------------
// GEMM1 + SiLU*mul + cw fold (double-buffered async-LDS pipeline):
//   actw[m, n] = cw[m,e] * silu(x@ws[e][n,:].T) * (x@ws[e][I+n,:].T)
// ---------------------------------------------------------------------------
__global__ __launch_bounds__(256) void moe_gemm1_silu_kernel(
    const float* __restrict__ x, const float* __restrict__ wse,
    const float* __restrict__ cw, int e, float* __restrict__ actw)
{
    extern __shared__ float smem[];
    // Buffer layout: [xs0][bs0][xs1][bs1]; select via runtime offset arithmetic
    // (no LDS-pointer arrays: their constant initializers don't link).
#define XS_BUF(b) (smem + (b) * (2 * TILE_F))
#define BS_BUF(b) (smem + TILE_F + (b) * (2 * TILE_F))

    const int tid  = threadIdx.x;
    const int wave = tid >> 5;
    const int lane = tid & 31;
    const int hi   = lane >> 4;
    const int lm   = lane & 15;
    const int m0   = blockIdx.x * BM;
    const int n0   = blockIdx.y * 64;     // act-column base

    v8f accg[4], accu[4];
#pragma unroll
    for (int t = 0; t < 4; ++t) { accg[t] = zero8(); accu[t] = zero8(); }

    // Stage issuer: 8 async b128 per thread (4 for x-tile + 4 for w-tile).
#define G1_ISSUE(k0, buf)                                                      \
    {                                                                          \
        float* xs_ = XS_BUF(buf);                                              \
        float* bs_ = BS_BUF(buf);                                              \
        _Pragma("unroll")                                                      \
        for (int it = 0; it < 4; ++it) {                                       \
            const int qi  = tid + it * 256;                                    \
            const int row = qi >> 3;                                           \
            const int seg = qi & 7;                                            \
            const float* gx = &x[(size_t)(m0 + row) * HID + (k0) + seg * 4];   \
            ASYNC_B128(lds_addr(&xs_[row * SX + seg * 4]), gx);                \
            const int gr = (row < 64) ? (n0 + row) : (INTER + n0 + (row - 64));\
            const float* gb = &wse[(size_t)gr * HID + (k0) + seg * 4];         \
            ASYNC_B128(lds_addr(&bs_[row * SX + seg * 4]), gb);                \
        }                                                                      \
    }

    G1_ISSUE(0, 0);
    int cur = 0;
    for (int k0 = 0; k0 < HID; k0 += BK) {
        if (k0 + BK < HID) {
            G1_ISSUE(k0 + BK, cur ^ 1);
            WAIT_ASYNC_8();     // current stage (8 older async ops) complete
        } else {
            WAIT_ASYNC_0();
        }
        __syncthreads();        // publish current tiles to all waves

        const float* xs_ = XS_BUF(cur);
        const float* bs_ = BS_BUF(cur);
        const float* arow = &xs_[(wave * 16 + lm) * SX + 2 * hi];
#pragma unroll
        for (int kk = 0; kk < BK; kk += 4) {
            v2f a = *(const v2f*)&arow[kk];
#pragma unroll
            for (int t = 0; t < 4; ++t) {
                v2f bg = *(const v2f*)&bs_[(t * 16 + lm) * SX + kk + 2 * hi];
                accg[t] = __builtin_amdgcn_wmma_f32_16x16x4_f32(
                    false, a, false, bg, (short)0, accg[t], false, false);
                v2f bu = *(const v2f*)&bs_[(64 + t * 16 + lm) * SX + kk + 2 * hi];
                accu[t] = __builtin_amdgcn_wmma_f32_16x16x4_f32(
                    false, a, false, bu, (short)0, accu[t], false, false);
            }
        }
        __syncthreads();        // all waves done with 'cur' before it is refilled
        cur ^= 1;
    }
#undef G1_ISSUE
#undef XS_BUF
#undef BS_BUF

    // Epilogue: act = silu(g)*u, scaled by cw[m, e]; C/D layout: (M=j+8*hi, N=lm)
#pragma unroll
    for (int t = 0; t < 4; ++t) {
#pragma unroll
        for (int j = 0; j < 8; ++j) {
            const int m = m0 + wave * 16 + j + 8 * hi;
            const float g = accg[t][j];
            const float u = accu[t][j];
            const float a = (g / (1.f + __expf(-g))) * u * cw[m * NE + e];
            actw[(size_t)m * INTER + n0 + t * 16 + lm] = a;
        }
    }
}

// ---------------------------------------------------------------------------
// GEMM2: out[m, n] (+)= sum_i actw[m, i] * w2s[e][n, i]  (same async pipeline)
// ---------------------------------------------------------------------------
__global__ __launch_bounds__(256) void moe_gemm2_kernel(
    const float* __restrict__ actw, const float* __restrict__ w2e,
    float* __restrict__ out, int first)
{
    extern __shared__ float smem[];
#define AS_BUF(b) (smem + (b) * (2 * TILE_F))
#define BS_BUF(b) (smem + TILE_F + (b) * (2 * TILE_F))

    const int tid  = threadIdx.x;
    const int wave = tid >> 5;
    const int lane = tid & 31;
    const int hi   = lane >> 4;
    const int lm   = lane & 15;
    const int m0   = blockIdx.x * BM;
    const int n0   = blockIdx.y * BM;

    v8f acc[8];
#pragma unroll
    for (int t = 0; t < 8; ++t) acc[t] = zero8();

#define G2_ISSUE(k0, buf)                                                      \
    {                                                                          \
        float* as_ = AS_BUF(buf);                                              \
        float* bs_ = BS_BUF(buf);                                              \
        _Pragma("unroll")                                                      \
        for (int it = 0; it < 4; ++it) {                                       \
            const int qi  = tid + it * 256;                                    \
            const int row = qi >> 3;                                           \
            const int seg = qi & 7;                                            \
            const float* ga = &actw[(size_t)(m0 + row) * INTER + (k0) + seg * 4]; \
            ASYNC_B128(lds_addr(&as_[row * SX + seg * 4]), ga);                \
            const float* gb = &w2e[(size_t)(n0 + row) * INTER + (k0) + seg * 4];  \
            ASYNC_B128(lds_addr(&bs_[row * SX + seg * 4]), gb);                \
        }                                                                      \
    }

    G2_ISSUE(0, 0);
    int cur = 0;
    for (int k0 = 0; k0 < INTER; k0 += BK) {
        if (k0 + BK < INTER) {
            G2_ISSUE(k0 + BK, cur ^ 1);
            WAIT_ASYNC_8();
        } else {
            WAIT_ASYNC_0();
        }
        __syncthreads();

        const float* as_ = AS_BUF(cur);
        const float* bs_ = BS_BUF(cur);
        const float* arow = &as_[(wave * 16 + lm) * SX + 2 * hi];
#pragma unroll
        for (int kk = 0; kk < BK; kk += 4) {
            v2f a = *(const v2f*)&arow[kk];
#pragma unroll
            for (int t = 0; t < 8; ++t) {
                v2f b = *(const v2f*)&bs_[(t * 16 + lm) * SX + kk + 2 * hi];
                acc[t] = __builtin_amdgcn_wmma_f32_16x16x4_f32(
                    false, a, false, b, (short)0, acc[t], false, false);
            }
        }
        __syncthreads();
        cur ^= 1;
    }
#undef G2_ISSUE
#undef AS_BUF
#undef BS_BUF

#pragma unroll
    for (int t = 0; t < 8; ++t) {
#pragma unroll
        for (int j = 0; j < 8; ++j) {
            const int m = m0 + wave * 16 + j + 8 * hi;
            const size_t idx = (size_t)m * HID + n0 + t * 16 + lm;
            const float v = acc[t][j];
            if (first) out[idx] = v;
            else       out[idx] += v;
        }
    }
}

// ---------------------------------------------------------------------------
extern "C" void kernel_launch(void* const* d_in, const int* in_sizes, int n_in,
                              void* d_out, int out_size, void* d_ws, size_t ws_size,
                              hipStream_t stream) {
    (void)in_sizes; (void)n_in; (void)out_size; (void)ws_size;
    const float* x   = (const float*)d_in[0];            // [T, H]
    const float* gw  = (const float*)d_in[1];            // [E, H]
    const float* ws  = (const float*)d_in[2];            // [E, 2I, H]
    const float* w2s = (const float*)d_in[3];            // [E, H, I]
    float* out = (float*)d_out;                          // [T, H]

    float* cw   = (float*)d_ws;                                        // [T, E]
    float* actw = (float*)((char*)d_ws + (size_t)T_TOK * NE * 4);      // [T, I]

    moe_router_kernel<<<T_TOK / 8, 256, 0, stream>>>(x, gw, cw);

    for (int e = 0; e < NE; ++e) {
        const float* wse = ws  + (size_t)e * 2 * INTER * HID;
        const float* w2e = w2s + (size_t)e * HID * INTER;
        moe_gemm1_silu_kernel<<<dim3(T_TOK / BM, INTER / 64), 256, SMEM_BYTES, stream>>>(
            x, wse, cw, e, actw);
        moe_gemm2_kernel<<<dim3(T_TOK / BM, HID / BM), 256, SMEM_BYTES, stream>>>(
            actw, w2e, out, (e == 0) ? 1 : 0);
    }
}